// GAT_74852690034750
// MI455X (gfx1250) — compile-verified
//
#include <hip/hip_runtime.h>
#include <hip/hip_bf16.h>
#include <math.h>

// ---------------------------------------------------------------------------
// GAT 2-layer forward for gfx1250 (MI455X).
//  - Layer-1 projection x@W1 on the matrix path: V_WMMA_F32_16X16X4_F32,
//    wave32, 16-row x 32-col tile per wave, 4 chained k-slices x 2 col tiles.
//  - Attention softmax: 3 edge passes per layer, recomputing the logit from
//    L2-resident [N,H] tables instead of materializing per-edge data.
//  - Segment max via monotone-uint mapping -> global_atomic_max_u32.
//  - Aggregation via global_atomic_add_f32 (L2 atomic units) with b128
//    feature gathers.
// ---------------------------------------------------------------------------

typedef __attribute__((ext_vector_type(2))) float v2f;
typedef __attribute__((ext_vector_type(8))) float v8f;

#define MONO_NEG_INF 0x007FFFFFu   // f2mono(-inf)

__device__ __forceinline__ unsigned f2mono(float f) {
  unsigned u = __float_as_uint(f);
  return (u & 0x80000000u) ? ~u : (u | 0x80000000u);
}
__device__ __forceinline__ float mono2f(unsigned m) {
  return (m & 0x80000000u) ? __uint_as_float(m ^ 0x80000000u)
                           : __uint_as_float(~m);
}
__device__ __forceinline__ float lrelu(float v) { return v >= 0.f ? v : 0.2f * v; }

// ---------------- fills ----------------
__global__ void k_fill_f32(float* p, long long n, float v) {
  long long i = (long long)blockIdx.x * blockDim.x + threadIdx.x;
  if (i < n) p[i] = v;
}
__global__ void k_fill_u32(unsigned* p, long long n, unsigned v) {
  long long i = (long long)blockIdx.x * blockDim.x + threadIdx.x;
  if (i < n) p[i] = v;
}

// ---------------- layer-1 projection: h1[N,32] = x[N,16] @ W1[16,32] --------
// One wave computes a 16x32 tile. f32 WMMA, K tiled 4x (16 = 4*4).
__global__ __launch_bounds__(256) void k_gemm1_wmma(
    const float* __restrict__ x, const float* __restrict__ W,
    float* __restrict__ h1, int N) {
  const int wavesPerBlock = blockDim.x >> 5;
  const int wave = blockIdx.x * wavesPerBlock + (threadIdx.x >> 5);
  const int lane = threadIdx.x & 31;
  const int row0 = wave * 16;
  if (row0 >= N) return;                 // whole-wave uniform exit

  const bool hi = lane >= 16;            // lanes 16..31 carry K+2 / M+8 halves
  const int l = lane & 15;
  int arow = row0 + l; if (arow >= N) arow = N - 1;   // safe (stores guarded)
  const float* xrow = x + (size_t)arow * 16;

  v8f c0 = {};  // cols 0..15
  v8f c1 = {};  // cols 16..31
#pragma unroll
  for (int s = 0; s < 4; ++s) {
    const int kr = 4 * s + (hi ? 2 : 0);
    v2f a;                               // A 16x4: lane holds K=kr, kr+1 of its row
    a.x = xrow[kr];
    a.y = xrow[kr + 1];
    v2f b0, b1;                          // B 4x16: row kr in vgpr0, kr+1 in vgpr1
    b0.x = W[kr * 32 + l];
    b0.y = W[(kr + 1) * 32 + l];
    b1.x = W[kr * 32 + 16 + l];
    b1.y = W[(kr + 1) * 32 + 16 + l];
    c0 = __builtin_amdgcn_wmma_f32_16x16x4_f32(false, a, false, b0, (short)0, c0, false, false);
    c1 = __builtin_amdgcn_wmma_f32_16x16x4_f32(false, a, false, b1, (short)0, c1, false, false);
  }
  // C layout: vgpr i holds M = i (+8 for hi lanes), N = lane&15
  float* base = h1 + (size_t)(row0 + (hi ? 8 : 0)) * 32 + l;
  if (row0 + 16 <= N) {                  // full tile: unguarded stores
#pragma unroll
    for (int i = 0; i < 8; ++i) {
      base[(size_t)i * 32]      = c0[i];
      base[(size_t)i * 32 + 16] = c1[i];
    }
  } else {                               // ragged tail (not hit for N%16==0)
#pragma unroll
    for (int i = 0; i < 8; ++i) {
      const int row = row0 + i + (hi ? 8 : 0);
      if (row < N) {
        base[(size_t)i * 32]      = c0[i];
        base[(size_t)i * 32 + 16] = c1[i];
      }
    }
  }
}

// ---------------- per-node attention logits (layer 1, H=4, C=8) -------------
__global__ void k_att1(const float* __restrict__ h1,
                       const float* __restrict__ attS, const float* __restrict__ attD,
                       float* __restrict__ aS, float* __restrict__ aD, int N) {
  const int i = blockIdx.x * blockDim.x + threadIdx.x;   // i = n*4 + head
  if (i >= N * 4) return;
  const int n = i >> 2, hh = i & 3;
  const float4* hp = (const float4*)(h1 + (size_t)n * 32 + hh * 8);
  const float4 h0 = hp[0], h1v = hp[1];
  const float4* as4 = (const float4*)(attS + hh * 8);
  const float4* ad4 = (const float4*)(attD + hh * 8);
  const float4 s0 = as4[0], s1 = as4[1], d0 = ad4[0], d1 = ad4[1];
  float s = h0.x * s0.x + h0.y * s0.y + h0.z * s0.z + h0.w * s0.w +
            h1v.x * s1.x + h1v.y * s1.y + h1v.z * s1.z + h1v.w * s1.w;
  float d = h0.x * d0.x + h0.y * d0.y + h0.z * d0.z + h0.w * d0.w +
            h1v.x * d1.x + h1v.y * d1.y + h1v.z * d1.z + h1v.w * d1.w;
  aS[i] = s;
  aD[i] = d;
}

// ---------------- edge helpers ----------------
__device__ __forceinline__ void edge_sd(const int* __restrict__ ei, long long e,
                                        int E, int N, int& s, int& d) {
  if (e < (long long)E) {
    s = ei[e];
    d = ei[(size_t)E + e];
    if ((unsigned)s >= (unsigned)N) s = 0;
    if ((unsigned)d >= (unsigned)N) d = 0;
  } else {                                // appended self-loops
    s = d = (int)(e - E);
  }
}

// pass 1: segment max of leaky_relu logits (monotone-uint atomic max)
template <int H>
__global__ void k_edge_max(const int* __restrict__ ei, int E, int N,
                           const float* __restrict__ aS, const float* __restrict__ aD,
                           unsigned* __restrict__ mx) {
  const long long e = (long long)blockIdx.x * blockDim.x + threadIdx.x;
  if (e >= (long long)E + N) return;
  int s, d; edge_sd(ei, e, E, N, s, d);
#pragma unroll
  for (int h = 0; h < H; ++h) {
    const float v = lrelu(aS[(size_t)s * H + h] + aD[(size_t)d * H + h]);
    atomicMax(&mx[(size_t)d * H + h], f2mono(v));
  }
}

// pass 2: denom[d] += exp(logit - max[d])
template <int H>
__global__ void k_edge_den(const int* __restrict__ ei, int E, int N,
                           const float* __restrict__ aS, const float* __restrict__ aD,
                           const unsigned* __restrict__ mx, float* __restrict__ den) {
  const long long e = (long long)blockIdx.x * blockDim.x + threadIdx.x;
  if (e >= (long long)E + N) return;
  int s, d; edge_sd(ei, e, E, N, s, d);
#pragma unroll
  for (int h = 0; h < H; ++h) {
    const float v = lrelu(aS[(size_t)s * H + h] + aD[(size_t)d * H + h]);
    atomicAdd(&den[(size_t)d * H + h], expf(v - mono2f(mx[(size_t)d * H + h])));
  }
}

// pass 3: out[d] += alpha * feat[s]   (one thread per (edge, head))
template <int H, int C>
__global__ void k_edge_acc(const int* __restrict__ ei, int E, int N,
                           const float* __restrict__ aS, const float* __restrict__ aD,
                           const unsigned* __restrict__ mx, const float* __restrict__ den,
                           const float* __restrict__ feat, float* __restrict__ out) {
  const long long t = (long long)blockIdx.x * blockDim.x + threadIdx.x;
  if (t >= ((long long)E + N) * H) return;
  const long long e = t / H;
  const int h = (int)(t - e * H);
  int s, d; edge_sd(ei, e, E, N, s, d);
  const float v = lrelu(aS[(size_t)s * H + h] + aD[(size_t)d * H + h]);
  const float alpha = expf(v - mono2f(mx[(size_t)d * H + h])) /
                      (den[(size_t)d * H + h] + 1e-16f);
  const float* fp = feat + ((size_t)s * H + h) * C;
  float* op = out + ((size_t)d * H + h) * C;
  if constexpr (C == 8) {
    // b128 gathers of the source feature row; scalar f32 atomics to dst
    const float4* f4 = (const float4*)fp;
    const float4 f0 = f4[0], f1 = f4[1];
    atomicAdd(&op[0], alpha * f0.x);
    atomicAdd(&op[1], alpha * f0.y);
    atomicAdd(&op[2], alpha * f0.z);
    atomicAdd(&op[3], alpha * f0.w);
    atomicAdd(&op[4], alpha * f1.x);
    atomicAdd(&op[5], alpha * f1.y);
    atomicAdd(&op[6], alpha * f1.z);
    atomicAdd(&op[7], alpha * f1.w);
  } else {
#pragma unroll
    for (int c = 0; c < C; ++c) atomicAdd(&op[c], alpha * fp[c]);
  }
}

// ---------------- layer-2 prep: elu(out1+b1) -> x2, h2 = x2@W2, logits ------
__global__ void k_layer2_prep(float* __restrict__ out1, const float* __restrict__ b1,
                              const float* __restrict__ W2,
                              const float* __restrict__ attS2, const float* __restrict__ attD2,
                              float* __restrict__ h2, float* __restrict__ aS2,
                              float* __restrict__ aD2, int N) {
  const int n = blockIdx.x * blockDim.x + threadIdx.x;
  if (n >= N) return;
  float* p = out1 + (size_t)n * 32;
  float acc = 0.f;
#pragma unroll
  for (int k = 0; k < 32; ++k) {
    float v = p[k] + b1[k];
    v = v > 0.f ? v : (expf(v) - 1.f);   // elu(alpha=1)
    p[k] = v;                             // x2 stored in place (ws)
    acc += v * W2[k];
  }
  h2[n] = acc;
  aS2[n] = acc * attS2[0];
  aD2[n] = acc * attD2[0];
}

// ---------------- final: out = out2 + b2 (mean over H=1 is identity) --------
__global__ void k_final(const float* __restrict__ out2, const float* __restrict__ b2,
                        float* __restrict__ out, int N) {
  const int n = blockIdx.x * blockDim.x + threadIdx.x;
  if (n < N) out[n] = out2[n] + b2[0];
}

// ---------------------------------------------------------------------------
extern "C" void kernel_launch(void* const* d_in, const int* in_sizes, int n_in,
                              void* d_out, int out_size, void* d_ws, size_t ws_size,
                              hipStream_t stream) {
  const float* x      = (const float*)d_in[0];
  const int*   ei     = (const int*)  d_in[1];   // [2, E] row-major: src row, dst row
  const float* W1     = (const float*)d_in[2];
  const float* attS1  = (const float*)d_in[3];
  const float* attD1  = (const float*)d_in[4];
  const float* b1     = (const float*)d_in[5];
  const float* W2     = (const float*)d_in[6];
  const float* attS2  = (const float*)d_in[7];
  const float* attD2  = (const float*)d_in[8];
  const float* b2     = (const float*)d_in[9];
  float* out = (float*)d_out;

  const int N = in_sizes[0] / 16;     // F_in = 16
  const int E = in_sizes[1] / 2;      // [2, E]
  const long long M = (long long)E + N;   // edges incl. self-loops

  // workspace carve-out (floats)
  float* ws = (float*)d_ws;
  float*    h1   = ws;                 ws += (size_t)N * 32;
  float*    aS1  = ws;                 ws += (size_t)N * 4;
  float*    aD1  = ws;                 ws += (size_t)N * 4;
  unsigned* mx1  = (unsigned*)ws;      ws += (size_t)N * 4;
  float*    den1 = ws;                 ws += (size_t)N * 4;
  float*    out1 = ws;                 ws += (size_t)N * 32;   // becomes x2
  float*    h2   = ws;                 ws += N;
  float*    aS2  = ws;                 ws += N;
  float*    aD2  = ws;                 ws += N;
  unsigned* mx2  = (unsigned*)ws;      ws += N;
  float*    den2 = ws;                 ws += N;
  float*    out2 = ws;                 ws += N;

  const int B = 256;
  auto blocks = [&](long long n) { return (unsigned)((n + B - 1) / B); };

  // init accumulators (every call: harness does not re-poison between replays)
  k_fill_f32<<<blocks((long long)N * 32), B, 0, stream>>>(out1, (long long)N * 32, 0.f);
  k_fill_u32<<<blocks((long long)N * 4), B, 0, stream>>>(mx1, (long long)N * 4, MONO_NEG_INF);
  k_fill_f32<<<blocks((long long)N * 4), B, 0, stream>>>(den1, (long long)N * 4, 0.f);
  k_fill_u32<<<blocks(N), B, 0, stream>>>(mx2, N, MONO_NEG_INF);
  k_fill_f32<<<blocks(N), B, 0, stream>>>(den2, N, 0.f);
  k_fill_f32<<<blocks(N), B, 0, stream>>>(out2, N, 0.f);

  // layer 1: projection (WMMA) + logits
  {
    const int wavesNeeded = (N + 15) / 16;
    const int wavesPerBlock = B / 32;
    k_gemm1_wmma<<<(wavesNeeded + wavesPerBlock - 1) / wavesPerBlock, B, 0, stream>>>(x, W1, h1, N);
  }
  k_att1<<<blocks((long long)N * 4), B, 0, stream>>>(h1, attS1, attD1, aS1, aD1, N);

  // layer 1: softmax-aggregate over edges (H=4, C=8)
  k_edge_max<4><<<blocks(M), B, 0, stream>>>(ei, E, N, aS1, aD1, mx1);
  k_edge_den<4><<<blocks(M), B, 0, stream>>>(ei, E, N, aS1, aD1, mx1, den1);
  k_edge_acc<4, 8><<<blocks(M * 4), B, 0, stream>>>(ei, E, N, aS1, aD1, mx1, den1, h1, out1);

  // layer 2 prep: elu + bias, projection to scalar, logits
  k_layer2_prep<<<blocks(N), B, 0, stream>>>(out1, b1, W2, attS2, attD2, h2, aS2, aD2, N);

  // layer 2: softmax-aggregate (H=1, C=1)
  k_edge_max<1><<<blocks(M), B, 0, stream>>>(ei, E, N, aS2, aD2, mx2);
  k_edge_den<1><<<blocks(M), B, 0, stream>>>(ei, E, N, aS2, aD2, mx2, den2);
  k_edge_acc<1, 1><<<blocks(M), B, 0, stream>>>(ei, E, N, aS2, aD2, mx2, den2, h2, out2);

  // epilogue
  k_final<<<blocks(N), B, 0, stream>>>(out2, b2, out, N);
  (void)n_in; (void)out_size; (void)ws_size;
}